// rnn_7395933684453
// MI455X (gfx1250) — compile-verified
//
#include <hip/hip_runtime.h>
#include <hip/hip_bf16.h>

// Problem constants (from reference)
#define BB 2048
#define TT 1024
#define II 28
#define HH 2
#define CC 10
#define NG 8            // 4*H gates
#define MM (BB * TT)    // GEMM M dimension

typedef __attribute__((ext_vector_type(2))) float v2f;
typedef __attribute__((ext_vector_type(8))) float v8f;

// ---------------------------------------------------------------------------
// Activation helpers: use gfx1250 v_tanh_f32 when the builtin exists.
// ---------------------------------------------------------------------------
__device__ __forceinline__ float fast_tanh(float x) {
#if __has_builtin(__builtin_amdgcn_tanhf)
    return __builtin_amdgcn_tanhf(x);
#else
    float e = __expf(-2.0f * x);
    return (1.0f - e) / (1.0f + e);
#endif
}
__device__ __forceinline__ float fast_sigm(float x) {
    return 0.5f + 0.5f * fast_tanh(0.5f * x);
}

// ---------------------------------------------------------------------------
// Phase 1: xg[t*B*8 + b*8 + g] = sum_i x[b,t,i]*W_ih[g,i] + b_ih[g] + b_hh[g]
// GEMM [M=B*T, K=28] x [K=28, N=16(pad)] via V_WMMA_F32_16X16X4_F32, K-loop=7.
// Bias folded into the C accumulator (broadcast per column).
// Output layout transposed to [T, B, 8] for coalesced scan-phase loads.
// ---------------------------------------------------------------------------
__global__ void lstm_xproj_wmma(const float* __restrict__ x,
                                const float* __restrict__ W_ih,
                                const float* __restrict__ b_ih,
                                const float* __restrict__ b_hh,
                                float* __restrict__ xg) {
    const int lane  = threadIdx.x & 31;
    const int n     = lane & 15;   // output column (gate index, 0..15; >=8 padding)
    const int khalf = lane >> 4;   // 0 -> K pair {0,1}; 1 -> K pair {2,3}

    // B fragments: B[k][n] = (n<8) ? W_ih[n*28 + k] : 0   (W_ih is [8,28] row-major)
    v2f bfrag[7];
#pragma unroll
    for (int j = 0; j < 7; ++j) {
        const int k0 = 4 * j + 2 * khalf;
        float b0 = 0.0f, b1 = 0.0f;
        if (n < NG) {
            b0 = W_ih[n * II + k0];
            b1 = W_ih[n * II + k0 + 1];
        }
        v2f bf; bf.x = b0; bf.y = b1;
        bfrag[j] = bf;
    }

    const float bias = (n < NG) ? (b_ih[n] + b_hh[n]) : 0.0f;
    v8f cbias;
#pragma unroll
    for (int v = 0; v < 8; ++v) cbias[v] = bias;

    const int wavesPerBlock = blockDim.x >> 5;
    const int gwave  = blockIdx.x * wavesPerBlock + (threadIdx.x >> 5);
    const int nwaves = gridDim.x * wavesPerBlock;
    const int ntiles = MM >> 4;   // 16 rows per tile

    for (int tile = gwave; tile < ntiles; tile += nwaves) {
        // A fragment rows: both lane halves load the same 16 rows, different K.
        const int   row  = tile * 16 + n;            // (b*T + t) flat row
        const float* xrow = x + (long)row * II;

        v8f c = cbias;
#pragma unroll
        for (int j = 0; j < 7; ++j) {
            const int k0 = 4 * j + 2 * khalf;
            v2f a; a.x = xrow[k0]; a.y = xrow[k0 + 1];
            // D = A(16x4) * B(4x16) + C  -> v_wmma_f32_16x16x4_f32
            c = __builtin_amdgcn_wmma_f32_16x16x4_f32(
                    false, a, false, bfrag[j], (short)0, c, false, false);
        }

        // D layout: VGPR v holds row m = v + 8*khalf, column n = lane&15.
        if (n < NG) {
#pragma unroll
            for (int v = 0; v < 8; ++v) {
                const int m  = v + 8 * khalf;
                const int r  = tile * 16 + m;        // r = b*T + t
                const int bi = r >> 10;              // / T (T = 1024)
                const int ti = r & (TT - 1);         // % T
                xg[((long)ti * BB + bi) * NG + n] = c[v];
            }
        }
    }
}

// ---------------------------------------------------------------------------
// Phase 2: sequential LSTM scan, one lane per batch element (latency-bound).
// Reads precomputed gates from xg[T,B,8]; double-buffered float4 prefetch.
// ---------------------------------------------------------------------------
__global__ void lstm_scan(const float* __restrict__ xg,
                          const float* __restrict__ W_hh,
                          const float* __restrict__ fc_W,
                          const float* __restrict__ fc_b,
                          float* __restrict__ out) {
    const int b = blockIdx.x * blockDim.x + threadIdx.x;
    if (b >= BB) return;

    // W_hh [8,2] row-major: uniform -> scalar loads
    float w[16];
#pragma unroll
    for (int k = 0; k < 16; ++k) w[k] = W_hh[k];

    const float4* g4 = (const float4*)xg;
    const long stride = (long)BB * 2;   // float4 units per timestep
    long idx = (long)b * 2;

    float h0 = 0.f, h1 = 0.f, c0 = 0.f, c1 = 0.f;
    float4 lo = g4[idx], hi = g4[idx + 1];

    for (int t = 0; t < TT; ++t) {
        const long nidx = idx + stride;
        float4 nlo = lo, nhi = hi;
        if (t + 1 < TT) { nlo = g4[nidx]; nhi = g4[nidx + 1]; }

        // torch gate order: i, f, g, o (each size H=2)
        const float i0 = lo.x + h0 * w[0]  + h1 * w[1];
        const float i1 = lo.y + h0 * w[2]  + h1 * w[3];
        const float f0 = lo.z + h0 * w[4]  + h1 * w[5];
        const float f1 = lo.w + h0 * w[6]  + h1 * w[7];
        const float g0 = hi.x + h0 * w[8]  + h1 * w[9];
        const float g1 = hi.y + h0 * w[10] + h1 * w[11];
        const float o0 = hi.z + h0 * w[12] + h1 * w[13];
        const float o1 = hi.w + h0 * w[14] + h1 * w[15];

        c0 = fast_sigm(f0) * c0 + fast_sigm(i0) * fast_tanh(g0);
        c1 = fast_sigm(f1) * c1 + fast_sigm(i1) * fast_tanh(g1);
        h0 = fast_sigm(o0) * fast_tanh(c0);
        h1 = fast_sigm(o1) * fast_tanh(c1);

        lo = nlo; hi = nhi; idx = nidx;
    }

    // Final FC: out[b,c] = h . fc_W[c,:] + fc_b[c]
#pragma unroll
    for (int c = 0; c < CC; ++c) {
        out[b * CC + c] = fc_b[c] + h0 * fc_W[c * HH + 0] + h1 * fc_W[c * HH + 1];
    }
}

// ---------------------------------------------------------------------------
// Fallback (only if workspace is too small): fully fused, no WMMA.
// ---------------------------------------------------------------------------
__global__ void lstm_fused(const float* __restrict__ x,
                           const float* __restrict__ W_ih,
                           const float* __restrict__ W_hh,
                           const float* __restrict__ b_ih,
                           const float* __restrict__ b_hh,
                           const float* __restrict__ fc_W,
                           const float* __restrict__ fc_b,
                           float* __restrict__ out) {
    const int b = blockIdx.x * blockDim.x + threadIdx.x;
    if (b >= BB) return;

    float w[16];
#pragma unroll
    for (int k = 0; k < 16; ++k) w[k] = W_hh[k];
    float bias[NG];
#pragma unroll
    for (int g = 0; g < NG; ++g) bias[g] = b_ih[g] + b_hh[g];

    float h0 = 0.f, h1 = 0.f, c0 = 0.f, c1 = 0.f;
    const float* xb = x + (long)b * TT * II;

    for (int t = 0; t < TT; ++t) {
        const float* xr = xb + (long)t * II;
        float ga[NG];
#pragma unroll
        for (int g = 0; g < NG; ++g) {
            float s = bias[g];
            for (int i = 0; i < II; ++i) s += xr[i] * W_ih[g * II + i];
            ga[g] = s;
        }
        const float i0 = ga[0] + h0 * w[0]  + h1 * w[1];
        const float i1 = ga[1] + h0 * w[2]  + h1 * w[3];
        const float f0 = ga[2] + h0 * w[4]  + h1 * w[5];
        const float f1 = ga[3] + h0 * w[6]  + h1 * w[7];
        const float g0 = ga[4] + h0 * w[8]  + h1 * w[9];
        const float g1 = ga[5] + h0 * w[10] + h1 * w[11];
        const float o0 = ga[6] + h0 * w[12] + h1 * w[13];
        const float o1 = ga[7] + h0 * w[14] + h1 * w[15];

        c0 = fast_sigm(f0) * c0 + fast_sigm(i0) * fast_tanh(g0);
        c1 = fast_sigm(f1) * c1 + fast_sigm(i1) * fast_tanh(g1);
        h0 = fast_sigm(o0) * fast_tanh(c0);
        h1 = fast_sigm(o1) * fast_tanh(c1);
    }
#pragma unroll
    for (int c = 0; c < CC; ++c) {
        out[b * CC + c] = fc_b[c] + h0 * fc_W[c * HH + 0] + h1 * fc_W[c * HH + 1];
    }
}

// ---------------------------------------------------------------------------
extern "C" void kernel_launch(void* const* d_in, const int* in_sizes, int n_in,
                              void* d_out, int out_size, void* d_ws, size_t ws_size,
                              hipStream_t stream) {
    const float* x    = (const float*)d_in[0];
    const float* W_ih = (const float*)d_in[1];
    const float* W_hh = (const float*)d_in[2];
    const float* b_ih = (const float*)d_in[3];
    const float* b_hh = (const float*)d_in[4];
    const float* fc_W = (const float*)d_in[5];
    const float* fc_b = (const float*)d_in[6];
    float* out = (float*)d_out;

    const size_t need = (size_t)MM * NG * sizeof(float);   // 64 MB
    if (ws_size >= need) {
        float* xg = (float*)d_ws;
        // Phase 1: WMMA input projection, 2048 blocks x 8 waves, 8 tiles/wave.
        lstm_xproj_wmma<<<2048, 256, 0, stream>>>(x, W_ih, b_ih, b_hh, xg);
        // Phase 2: 64 single-wave blocks (one lane per batch chain) to spread
        // the 64 latency-bound waves across WGPs.
        lstm_scan<<<BB / 32, 32, 0, stream>>>(xg, W_hh, fc_W, fc_b, out);
    } else {
        lstm_fused<<<BB / 32, 32, 0, stream>>>(x, W_ih, W_hh, b_ih, b_hh,
                                               fc_W, fc_b, out);
    }
}